// PPISPSlangtorch_64407329571617
// MI455X (gfx1250) — compile-verified
//
#include <hip/hip_runtime.h>
#include <hip/hip_bf16.h>
#include <math.h>

// CDNA5 async global->LDS path (ASYNCcnt). Flip to 0 if assembler rejects.
#define USE_ASYNC_LDS 1

constexpr int NUM_CAMERAS = 6;
constexpr int NUM_FRAMES  = 1000;

// Derived-table layout in d_ws (float4-aligned, all offsets in float4 units):
//   [0     , 250  )  exp2(exposure)[1000]            (1000 floats)
//   [250   , 2250 )  color: {e^cp0,e^cp1,cp2..cp7} per frame (1000*8 floats)
//   [2250  , 2268 )  CRF: {a,b,c,d} post-softplus, (cam*3+ch)*4 (72 floats)
//   [2268  , 2304 )  vignetting padded to 8 floats per (cam,ch) (144 floats)
constexpr int COLP_V4 = 250;
constexpr int CRF_V4  = 2250;
constexpr int VIG_V4  = 2268;
constexpr int TAB_V4  = 2304;          // 36 KB total, 2304 = 9 * 256 exactly
constexpr int TAB_F   = TAB_V4 * 4;    // 9216 floats

constexpr int BLOCK = 256;             // 8 wave32 per workgroup
constexpr int PPT   = 32;              // points per thread -> 8192 pts/block

// 12-byte, 4-aligned vector so (N,3) rows can be moved as single b96 ops.
typedef float v3f __attribute__((ext_vector_type(3), aligned(4)));
typedef float v2f __attribute__((ext_vector_type(2), aligned(4)));

__device__ __forceinline__ float softplusf(float x) {
  return (x > 20.0f) ? x : log1pf(expf(x));
}

// ---------------------------------------------------------------------------
// Prep: build derived tables once (9216 threads total). Hoists ALL per-table
// transcendentals out of the 8.4M-point main loop.
// ---------------------------------------------------------------------------
__global__ __launch_bounds__(BLOCK) void ppisp_prep(
    const float* __restrict__ exposure,   // [1000]
    const float* __restrict__ vig,        // [6,3,5]
    const float* __restrict__ colp,       // [1000,8]
    const float* __restrict__ crf,        // [6,3,4]
    float* __restrict__ ws) {
  int tid = blockIdx.x * BLOCK + threadIdx.x;
  if (tid >= TAB_F) return;
  float v;
  if (tid < NUM_FRAMES) {                         // exp2(exposure)
    v = exp2f(exposure[tid]);
  } else if (tid < NUM_FRAMES + NUM_FRAMES * 8) { // color params (exp on 0,1)
    int i = tid - NUM_FRAMES;
    int j = i & 7;
    v = colp[i];
    if (j < 2) v = expf(v);
  } else if (tid < CRF_V4 * 4 + 72) {             // CRF softplus transform
    int i = tid - CRF_V4 * 4;
    int k = i & 3;
    float q = crf[i];
    v = (k == 3) ? q : (softplusf(q) + ((k == 2) ? 0.1f : 0.3f));
  } else {                                        // vignetting, padded 5 -> 8
    int i  = tid - VIG_V4 * 4;
    int pc = i >> 3;
    int k  = i & 7;
    v = (k < 5) ? vig[pc * 5 + k] : 0.0f;
  }
  ws[tid] = v;
}

// ---------------------------------------------------------------------------
// Main: stream 8.4M points; all gathers served from LDS (ds_load_b128).
// ---------------------------------------------------------------------------
__global__ __launch_bounds__(BLOCK) void ppisp_main(
    const float* __restrict__ ws,
    const float* __restrict__ rgb,
    const float* __restrict__ pcoord,
    const int*  __restrict__ cams,
    const int*  __restrict__ frms,
    const int*  __restrict__ resW,
    const int*  __restrict__ resH,
    float* __restrict__ out,
    int n) {
  __shared__ float4 s_tab4[TAB_V4];
  const float4* src4 = reinterpret_cast<const float4*>(ws);

#if USE_ASYNC_LDS
  // CDNA5 async copy engine: per-lane global->LDS B128 transfers tracked by
  // ASYNCcnt. 2304 = 9 * 256, so every lane is active on every iteration.
  for (int i = threadIdx.x; i < TAB_V4; i += BLOCK) {
    unsigned lds_off = (unsigned)(size_t)(&s_tab4[i]);
    unsigned long long gaddr = (unsigned long long)(size_t)(src4 + i);
    asm volatile("global_load_async_to_lds_b128 %0, %1, off"
                 :: "v"(lds_off), "v"(gaddr) : "memory");
  }
  asm volatile("s_wait_asynccnt 0" ::: "memory");
#else
  for (int i = threadIdx.x; i < TAB_V4; i += BLOCK) s_tab4[i] = src4[i];
#endif
  __syncthreads();

  const float*  s_expo  = reinterpret_cast<const float*>(s_tab4);
  const float4* s_colp4 = s_tab4 + COLP_V4;
  const float4* s_crf4  = s_tab4 + CRF_V4;
  const float4* s_vig4  = s_tab4 + VIG_V4;

  const v3f* __restrict__ rgb3 = reinterpret_cast<const v3f*>(rgb);
  const v2f* __restrict__ pc2  = reinterpret_cast<const v2f*>(pcoord);
  v3f* __restrict__ out3       = reinterpret_cast<v3f*>(out);

  const float invW = 1.0f / (float)resW[0];
  const float invH = 1.0f / (float)resH[0];

  const int base = blockIdx.x * (BLOCK * PPT) + threadIdx.x;

#pragma unroll 2
  for (int k = 0; k < PPT; ++k) {
    int idx = base + k * BLOCK;
    if (idx < n) {
      int cam = cams[idx];
      int frm = frms[idx];
      v2f pxy = pc2[idx];     // global_load_b64
      v3f rin = rgb3[idx];    // global_load_b96

      // exposure (pre-exponentiated)
      float e = s_expo[frm];
      float u = pxy.x * invW - 0.5f;
      float v = pxy.y * invH - 0.5f;

      float x[3] = {rin.x * e, rin.y * e, rin.z * e};
      int pcb = cam * 3;

      // vignetting per channel
#pragma unroll
      for (int ch = 0; ch < 3; ++ch) {
        float4 w   = s_vig4[(pcb + ch) * 2];        // vp0..vp3 (ds_load_b128)
        float  vp4 = s_vig4[(pcb + ch) * 2 + 1].x;  // vp4
        float du = u - w.x;
        float dv = v - w.y;
        float r2 = du * du + dv * dv;
        x[ch] *= 1.0f + r2 * (w.z + r2 * (w.w + r2 * vp4));
      }

      // color transform ({e^cp0, e^cp1} pre-exponentiated)
      float4 c0 = s_colp4[frm * 2 + 0];
      float4 c1 = s_colp4[frm * 2 + 1];
      float r = x[0] * c0.x;
      float g = x[1];
      float b = x[2] * c0.y;
      float o[3];
      o[0] = r + c0.z * g + c0.w * b;
      o[1] = c1.x * r + g + c1.y * b;
      o[2] = c1.z * r + c1.w * g + b;

      // CRF: y = ((1+c)*x^a / (x^a + c))^b + d   (a,b,c pre-softplus'd)
      v3f y;
#pragma unroll
      for (int ch = 0; ch < 3; ++ch) {
        float4 q  = s_crf4[pcb + ch];
        float xc  = fmaxf(o[ch], 1e-6f);
        float xa  = exp2f(q.x * __log2f(xc));
        float t   = ((1.0f + q.z) * xa) / (xa + q.z);
        y[ch]     = exp2f(q.y * __log2f(t)) + q.w;
      }
      // single non-temporal b96 store (write-once stream; keep L2 for tables)
      __builtin_nontemporal_store(y, &out3[idx]);
    }
  }
}

// ---------------------------------------------------------------------------
extern "C" void kernel_launch(void* const* d_in, const int* in_sizes, int n_in,
                              void* d_out, int out_size, void* d_ws, size_t ws_size,
                              hipStream_t stream) {
  const float* exposure = (const float*)d_in[0];
  const float* vig      = (const float*)d_in[1];
  const float* colp     = (const float*)d_in[2];
  const float* crf      = (const float*)d_in[3];
  const float* rgb      = (const float*)d_in[4];
  const float* pcoord   = (const float*)d_in[5];
  const int*   cams     = (const int*)d_in[6];
  const int*   frms     = (const int*)d_in[7];
  const int*   resW     = (const int*)d_in[8];
  const int*   resH     = (const int*)d_in[9];
  float*       out      = (float*)d_out;
  float*       ws       = (float*)d_ws;

  int n = in_sizes[4] / 3;  // rgb is (N,3)

  int prep_blocks = (TAB_F + BLOCK - 1) / BLOCK;  // 36
  ppisp_prep<<<prep_blocks, BLOCK, 0, stream>>>(exposure, vig, colp, crf, ws);

  int ppb = BLOCK * PPT;                          // 8192 points per block
  int blocks = (n + ppb - 1) / ppb;               // 1024 for N = 2^23
  ppisp_main<<<blocks, BLOCK, 0, stream>>>(ws, rgb, pcoord, cams, frms,
                                           resW, resH, out, n);
}